// LabelHandler_66477503808042
// MI455X (gfx1250) — compile-verified
//
#include <hip/hip_runtime.h>

typedef float v2f __attribute__((ext_vector_type(2)));
typedef float v8f __attribute__((ext_vector_type(8)));

#define MLP   512
#define EMBED 300
#define NB    4
#define NS    36
#define NV    2000

// C[M x N] = A[M x K] @ W[K x N], all row-major.
// Requires M%16==0, K%4==0, N%64==0. One wave per 16x64 output strip.
// Uses V_WMMA_F32_16X16X4_F32 (full fp32 precision on the matrix pipes).
__global__ __launch_bounds__(32) void gemm_f32_wmma(const float* __restrict__ A,
                                                    const float* __restrict__ W,
                                                    float* __restrict__ C,
                                                    int M, int K, int N) {
    const int lane = threadIdx.x;            // 0..31, wave32
    const int m0   = blockIdx.y * 16;
    const int n0   = blockIdx.x * 64;
    const int row  = m0 + (lane & 15);       // A-frag: M = lane%16
    const int kh   = (lane >> 4) * 2;        // A/B-frag: K-half select (0 or 2)
    const int col  = (lane & 15);            // B-frag: N = lane%16

    v8f acc0 = {}, acc1 = {}, acc2 = {}, acc3 = {};

    for (int k = 0; k < K; k += 4) {
        // A fragment: lane l -> A[row][k+kh], A[row][k+kh+1] (8B-aligned float2)
        v2f a = *(const v2f*)(A + (size_t)row * K + k + kh);

        const float* w0 = W + (size_t)(k + kh) * N;       // K-row k+kh
        const float* w1r = w0 + N;                        // K-row k+kh+1
        v2f b;
        b.x = w0[n0 +  0 + col]; b.y = w1r[n0 +  0 + col];
        acc0 = __builtin_amdgcn_wmma_f32_16x16x4_f32(false, a, false, b, (short)0, acc0, false, false);
        b.x = w0[n0 + 16 + col]; b.y = w1r[n0 + 16 + col];
        acc1 = __builtin_amdgcn_wmma_f32_16x16x4_f32(false, a, false, b, (short)0, acc1, false, false);
        b.x = w0[n0 + 32 + col]; b.y = w1r[n0 + 32 + col];
        acc2 = __builtin_amdgcn_wmma_f32_16x16x4_f32(false, a, false, b, (short)0, acc2, false, false);
        b.x = w0[n0 + 48 + col]; b.y = w1r[n0 + 48 + col];
        acc3 = __builtin_amdgcn_wmma_f32_16x16x4_f32(false, a, false, b, (short)0, acc3, false, false);
    }

    // D layout: VGPR j -> row m0 + j + 8*(lane/16), col n0 + lane%16
    const int mrow = m0 + (lane >> 4) * 8;
    const int ncol = n0 + (lane & 15);
#pragma unroll
    for (int j = 0; j < 8; ++j) {
        float* cr = C + (size_t)(mrow + j) * N + ncol;
        cr[ 0] = acc0[j];
        cr[16] = acc1[j];
        cr[32] = acc2[j];
        cr[48] = acc3[j];
    }
}

// One wave32 per (b, v): fused bias-relu-dot logits (never materializes
// the B*V*S*MLP "joint" tensor), 36-wide softmax in registers, then
// attention-weighted accumulation of img into v_emb. Lane i owns channels
// h = i*32 + lane (stride-32 => coalesced loads/stores).
__global__ __launch_bounds__(32) void attn_softmax_kernel(const float* __restrict__ hv,
                                                          const float* __restrict__ hq,
                                                          const float* __restrict__ img,
                                                          const float* __restrict__ b1,
                                                          const float* __restrict__ w2,
                                                          float* __restrict__ out) {
    const int v    = blockIdx.x;
    const int b    = blockIdx.y;
    const int lane = threadIdx.x;

    float hqv[16], bia[16], wq[16];
#pragma unroll
    for (int i = 0; i < 16; ++i) {
        const int h = i * 32 + lane;
        hqv[i] = hq[(size_t)v * MLP + h];
        bia[i] = b1[h];
        wq[i]  = w2[h];
    }

    const float* hvb = hv + (size_t)b * NS * MLP;

    // Pass 1: logits[s] = sum_h relu(hv + hq + b1) * w2   (b2 cancels in softmax)
    float logit[NS];
    float mx = -3.4e38f;
    for (int s = 0; s < NS; ++s) {
        const float* hr = hvb + s * MLP;
        float acc = 0.f;
#pragma unroll
        for (int i = 0; i < 16; ++i) {
            float x = hr[i * 32 + lane] + hqv[i] + bia[i];
            x = fmaxf(x, 0.f);
            acc = fmaf(x, wq[i], acc);
        }
#pragma unroll
        for (int off = 16; off > 0; off >>= 1)
            acc += __shfl_xor(acc, off, 32);
        logit[s] = acc;
        mx = fmaxf(mx, acc);
    }

    // Softmax over S
    float denom = 0.f;
#pragma unroll
    for (int s = 0; s < NS; ++s) {
        const float e = __expf(logit[s] - mx);
        logit[s] = e;
        denom += e;
    }
    const float inv = 1.0f / denom;

    // Pass 2: v_emb[h] = sum_s att[s] * img[b,s,h]
    const float* imgb = img + (size_t)b * NS * MLP;
    float vacc[16] = {};
    for (int s = 0; s < NS; ++s) {
        const float a = logit[s] * inv;
        const float* ir = imgb + s * MLP;
#pragma unroll
        for (int i = 0; i < 16; ++i)
            vacc[i] = fmaf(a, ir[i * 32 + lane], vacc[i]);
    }

    float* o = out + ((size_t)b * NV + v) * MLP;
#pragma unroll
    for (int i = 0; i < 16; ++i)
        o[i * 32 + lane] = vacc[i];
}

extern "C" void kernel_launch(void* const* d_in, const int* in_sizes, int n_in,
                              void* d_out, int out_size, void* d_ws, size_t ws_size,
                              hipStream_t stream) {
    const float* img          = (const float*)d_in[0];  // (B, S, MLP)
    const float* label_weight = (const float*)d_in[1];  // (V, EMBED)
    const float* w1           = (const float*)d_in[2];  // (MLP+EMBED, MLP)
    const float* b1           = (const float*)d_in[3];  // (MLP,)
    const float* w2           = (const float*)d_in[4];  // (MLP, 1)
    // d_in[5] = b2: constant shift of logits, cancels under softmax(axis=S).
    float* out = (float*)d_out;                         // (B*V, MLP)

    float* hv = (float*)d_ws;                 // (B*S, MLP)  = 73728 floats
    float* hq = hv + (size_t)NB * NS * MLP;   // (V, MLP)    = 1024000 floats

    // hv = img(144x512) @ w1[:512](512x512)
    gemm_f32_wmma<<<dim3(MLP / 64, (NB * NS) / 16), 32, 0, stream>>>(
        img, w1, hv, NB * NS, MLP, MLP);

    // hq = label_weight(2000x300) @ w1[512:](300x512)
    gemm_f32_wmma<<<dim3(MLP / 64, NV / 16), 32, 0, stream>>>(
        label_weight, w1 + (size_t)MLP * MLP, hq, NV, EMBED, MLP);

    // Fused joint/relu/logits/softmax/weighted-sum
    attn_softmax_kernel<<<dim3(NV, NB), 32, 0, stream>>>(
        hv, hq, img, b1, w2, out);
}